// myGCN_87462714016644
// MI455X (gfx1250) — compile-verified
//
#include <hip/hip_runtime.h>

// ---------------------------------------------------------------------------
// GCN forward for MI455X (gfx1250): fp32 WMMA (v_wmma_f32_16x16x4_f32) for the
// dense node GEMMs, atomic scatter for edges, self-loops folded into a
// non-atomic init. Entire working set (~52MB) is L2-resident on MI455X.
// Round 2: hoisted tile-uniform bounds checks out of the WMMA store/load path
// (fast full-tile path = unguarded b32 stores, no per-element exec juggling).
// ---------------------------------------------------------------------------

typedef __attribute__((ext_vector_type(2))) float v2f;
typedef __attribute__((ext_vector_type(8))) float v8f;

#define DFEAT 128
#define DOUT  40

// ---------------- degree / norm ----------------
__global__ void k_deg_init(float* __restrict__ deg, int N) {
    int i = blockIdx.x * blockDim.x + threadIdx.x;
    if (i < N) deg[i] = 1.0f;  // self-loop contributes 1 to every node
}

__global__ void k_deg_accum(const int* __restrict__ dst, float* __restrict__ deg, int E) {
    int e = blockIdx.x * blockDim.x + threadIdx.x;
    if (e < E) atomicAdd(&deg[dst[e]], 1.0f);
}

__global__ void k_dinv(const float* __restrict__ deg, float* __restrict__ dinv, int N) {
    int i = blockIdx.x * blockDim.x + threadIdx.x;
    if (i < N) dinv[i] = rsqrtf(deg[i]);  // deg >= 1 always (self-loops)
}

// ---------------- dense GEMM: Y[M x 128] = X[M x 128] @ W[128 x 128] -------
// 256 threads = 8 waves; block owns a 16-row strip, wave w owns cols [16w,16w+16).
// A strip staged in LDS (stride 132 floats -> 4-bank rotation, conflict-free).
// WMMA f32 16x16x4: A lanes 0-15 = M rows (K pair 0,1), lanes 16-31 same rows
// (K pair 2,3); B/C/D rows striped across lanes per VGPR (ISA 7.12.2).
__global__ void k_gemm_wmma(const float* __restrict__ X, const float* __restrict__ W,
                            float* __restrict__ Y, int M) {
    __shared__ float As[16 * 132];
    const int row0 = blockIdx.x * 16;
    const int tid  = threadIdx.x;
    const bool full = (row0 + 16 <= M);  // uniform per block

    if (full) {
        // 16*128 floats, 256 threads -> 8 each, no guards
#pragma unroll
        for (int it = 0; it < 8; ++it) {
            int idx = tid + it * 256;
            int r = idx >> 7, c = idx & 127;
            As[r * 132 + c] = X[row0 * DFEAT + idx];
        }
    } else {
        for (int idx = tid; idx < 16 * DFEAT; idx += 256) {
            int r = idx >> 7, c = idx & 127;
            int gr = row0 + r;
            As[r * 132 + c] = (gr < M) ? X[gr * DFEAT + c] : 0.0f;
        }
    }
    __syncthreads();

    const int wave = tid >> 5;
    const int lane = tid & 31;
    const int n0   = wave * 16;
    const int lm   = lane & 15;          // M row (A) / N col (B,C,D)
    const int kh   = (lane >> 4) << 1;   // K sub-pair: 0 or 2

    v8f acc = {};
#pragma unroll 4
    for (int k = 0; k < DFEAT; k += 4) {
        v2f a, b;
        a.x = As[lm * 132 + k + kh];
        a.y = As[lm * 132 + k + kh + 1];
        b.x = W[(k + kh) * DFEAT + n0 + lm];
        b.y = W[(k + kh + 1) * DFEAT + n0 + lm];
        acc = __builtin_amdgcn_wmma_f32_16x16x4_f32(
            /*neg_a=*/false, a, /*neg_b=*/false, b,
            /*c_mod=*/(short)0, acc, /*reuse_a=*/false, /*reuse_b=*/false);
    }

    // C/D layout: VGPR v -> M = v + (lane<16 ? 0 : 8), N = n0 + lm
    const int mbase = row0 + ((lane >> 4) << 3);
    float* yp = Y + (size_t)mbase * DFEAT + n0 + lm;
    if (full) {
#pragma unroll
        for (int v = 0; v < 8; ++v) yp[v * DFEAT] = acc[v];
    } else {
#pragma unroll
        for (int v = 0; v < 8; ++v) {
            if (mbase + v < M) yp[v * DFEAT] = acc[v];
        }
    }
}

// ---------------- self-loop init: agg = h * dinv^2 (no atomics) ------------
__global__ void k_self_init(const float* __restrict__ h, const float* __restrict__ dinv,
                            float* __restrict__ agg, int N) {
    int idx = blockIdx.x * blockDim.x + threadIdx.x;  // over N*32 float4s
    if (idx >= N * (DFEAT / 4)) return;
    int i = idx >> 5;
    float w = dinv[i] * dinv[i];
    float4 v = ((const float4*)h)[idx];
    v.x *= w; v.y *= w; v.z *= w; v.w *= w;
    ((float4*)agg)[idx] = v;
}

// ---------------- edge scatter: agg[dst] += h[src] * dinv[src]*dinv[dst] ---
__global__ void k_scatter(const float* __restrict__ h, const int* __restrict__ src,
                          const int* __restrict__ dst, const float* __restrict__ dinv,
                          float* __restrict__ agg, int E) {
    int e = blockIdx.x * 2 + (threadIdx.x >> 7);
    int d = threadIdx.x & 127;
    if (e >= E) return;
    int s = src[e], t = dst[e];
    float w = dinv[s] * dinv[t];
    atomicAdd(&agg[t * DFEAT + d], h[s * DFEAT + d] * w);
}

// ---------------- bias + relu in place -------------------------------------
__global__ void k_bias_relu(float* __restrict__ a, const float* __restrict__ b, int N) {
    int idx = blockIdx.x * blockDim.x + threadIdx.x;
    if (idx >= N * DFEAT) return;
    float v = a[idx] + b[idx & 127];
    a[idx] = v > 0.0f ? v : 0.0f;
}

// ---------------- pooling --------------------------------------------------
__global__ void k_zero(float* __restrict__ p, int n) {
    int i = blockIdx.x * blockDim.x + threadIdx.x;
    if (i < n) p[i] = 0.0f;
}

__global__ void k_count(const int* __restrict__ batch, float* __restrict__ counts, int N) {
    int i = blockIdx.x * blockDim.x + threadIdx.x;
    if (i < N) atomicAdd(&counts[batch[i]], 1.0f);
}

__global__ void k_pool_sum(const float* __restrict__ h, const int* __restrict__ batch,
                           float* __restrict__ pooled, int N) {
    int idx = blockIdx.x * blockDim.x + threadIdx.x;
    if (idx >= N * DFEAT) return;
    int i = idx >> 7, d = idx & 127;
    atomicAdd(&pooled[batch[i] * DFEAT + d], h[idx]);
}

// ---------------- head: out[g] = (pooled[g]/cnt + b2) @ linW + linb --------
__global__ void k_head(const float* __restrict__ pooled, const float* __restrict__ counts,
                       const float* __restrict__ b2, const float* __restrict__ linW,
                       const float* __restrict__ linb, float* __restrict__ out) {
    __shared__ float mean[DFEAT];
    int g = blockIdx.x;
    int t = threadIdx.x;  // blockDim = 128
    float cnt = counts[g];
    float m = 0.0f;
    if (cnt > 0.0f) m = pooled[g * DFEAT + t] / cnt + b2[t];
    mean[t] = m;
    __syncthreads();
    if (t < DOUT) {
        float acc = linb[t];
#pragma unroll 8
        for (int d = 0; d < DFEAT; ++d) acc += mean[d] * linW[d * DOUT + t];
        out[g * DOUT + t] = acc;
    }
}

// ---------------------------------------------------------------------------
extern "C" void kernel_launch(void* const* d_in, const int* in_sizes, int n_in,
                              void* d_out, int out_size, void* d_ws, size_t ws_size,
                              hipStream_t stream) {
    const float* x    = (const float*)d_in[0];
    const float* W1   = (const float*)d_in[1];
    const float* b1   = (const float*)d_in[2];
    const float* W2   = (const float*)d_in[3];
    const float* b2   = (const float*)d_in[4];
    const float* linW = (const float*)d_in[5];
    const float* linb = (const float*)d_in[6];
    const int*   eidx = (const int*)d_in[7];   // [2, E] row-major: src then dst
    const int*   batch = (const int*)d_in[8];
    float* out = (float*)d_out;

    const int N = in_sizes[0] / DFEAT;   // 50000
    const int E = in_sizes[7] / 2;       // 600000
    const int G = out_size / DOUT;       // 100
    const int* esrc = eidx;
    const int* edst = eidx + E;

    // workspace layout (floats)
    float* ws     = (float*)d_ws;
    float* deg    = ws;                          // N
    float* dinv   = deg + N;                     // N
    float* bufA   = dinv + N;                    // N*128
    float* bufB   = bufA + (size_t)N * DFEAT;    // N*128
    float* pooled = bufB + (size_t)N * DFEAT;    // G*128
    float* counts = pooled + (size_t)G * DFEAT;  // G

    const int T = 256;
    dim3 blk(T);
    int gN    = (N + T - 1) / T;
    int gE    = (E + T - 1) / T;
    int gNF   = ((size_t)N * DFEAT + T - 1) / T;
    int gNF4  = ((size_t)N * (DFEAT / 4) + T - 1) / T;
    int gTile = (N + 15) / 16;
    int gEdge = (E + 1) / 2;  // 2 edges per 256-thread block

    // normalization
    k_deg_init<<<gN, blk, 0, stream>>>(deg, N);
    k_deg_accum<<<gE, blk, 0, stream>>>(edst, deg, E);
    k_dinv<<<gN, blk, 0, stream>>>(deg, dinv, N);

    // layer 1: h0 = x @ W1 ; agg = D^-1/2 A D^-1/2 h0 ; relu(agg + b1)
    k_gemm_wmma<<<gTile, blk, 0, stream>>>(x, W1, bufA, N);
    k_self_init<<<gNF4, blk, 0, stream>>>(bufA, dinv, bufB, N);
    k_scatter<<<gEdge, blk, 0, stream>>>(bufA, esrc, edst, dinv, bufB, E);
    k_bias_relu<<<gNF, blk, 0, stream>>>(bufB, b1, N);

    // layer 2: h1 = h @ W2 ; agg2 (bias b2 folded into head)
    k_gemm_wmma<<<gTile, blk, 0, stream>>>(bufB, W2, bufA, N);
    k_self_init<<<gNF4, blk, 0, stream>>>(bufA, dinv, bufB, N);
    k_scatter<<<gEdge, blk, 0, stream>>>(bufA, esrc, edst, dinv, bufB, E);

    // global mean pool
    int gZ = (G * DFEAT + G + T - 1) / T;
    k_zero<<<gZ, blk, 0, stream>>>(pooled, G * DFEAT + G);  // pooled + counts contiguous
    k_count<<<gN, blk, 0, stream>>>(batch, counts, N);
    k_pool_sum<<<gNF, blk, 0, stream>>>(bufB, batch, pooled, N);

    // head
    k_head<<<G, dim3(DFEAT), 0, stream>>>(pooled, counts, b2, linW, linb, out);
}